// AttentionPooling_21251498180614
// MI455X (gfx1250) — compile-verified
//
#include <hip/hip_runtime.h>
#include <math.h>

// ---------------------------------------------------------------------------
// AttentionPooling, restructured:
//   scores = (x @ qk) * scale                where qk = query @ Wk  (bk cancels)
//   attn   = segment_softmax(scores)
//   s[g]   = sum_n attn[n] * x[n]            (weighted segment sum, sum attn = 1)
//   out    = s @ Wc.T + bc                   where Wc = Wo@Wv, bc = Wo@bv + bo
// Dominant cost: one streaming pass over x (256 MB) -> bandwidth bound.
// Final GEMM uses V_WMMA_F32_16X16X4_F32 (full fp32 precision).
// ---------------------------------------------------------------------------

typedef float v2f __attribute__((ext_vector_type(2)));
typedef float v8f __attribute__((ext_vector_type(8)));

#define HID 128          // hidden size (fixed by the problem)
#define SCAP 3072        // max scores cached in LDS per graph (fallback to global beyond)

// qk[t] = sum_i query[i] * Wk[i*H+t];  bc[t] = sum_j Wo[t*H+j]*bv[j] + bo[t]
__global__ void setup_small(const float* __restrict__ query, const float* __restrict__ Wk,
                            const float* __restrict__ Wo, const float* __restrict__ bv,
                            const float* __restrict__ bo,
                            float* __restrict__ qk, float* __restrict__ bc) {
  int t = threadIdx.x;  // 0..127
  float accq = 0.f;
  for (int i = 0; i < HID; ++i) accq = fmaf(query[i], Wk[i * HID + t], accq);
  qk[t] = accq;
  float accb = 0.f;
  for (int j = 0; j < HID; ++j) accb = fmaf(Wo[t * HID + j], bv[j], accb);
  bc[t] = accb + bo[t];
}

// Wc[n][k] = sum_j Wo[n][j] * Wv[j][k]   (block n, thread k; Wv reads coalesced)
__global__ void setup_wc(const float* __restrict__ Wo, const float* __restrict__ Wv,
                         float* __restrict__ Wc) {
  int n = blockIdx.x, k = threadIdx.x;
  float acc = 0.f;
  for (int j = 0; j < HID; ++j) acc = fmaf(Wo[n * HID + j], Wv[j * HID + k], acc);
  Wc[n * HID + k] = acc;
}

// One block (128 threads = 4 waves) per graph segment.
// Phase 1: scores (wave-per-node GEMV, wave32 shuffle reduce) -> LDS (or global fallback)
// Phase 2: block max; Phase 3: exp + block sum; Phase 4: weighted sum (thread = column)
__global__ void graph_pool(const float* __restrict__ x, const int* __restrict__ batch,
                           const float* __restrict__ qk, float scale,
                           float* __restrict__ attnbuf, float* __restrict__ sbuf,
                           int N, int G) {
  __shared__ float red[128];
  __shared__ float sc[SCAP];
  __shared__ int seg[2];
  const int g = blockIdx.x;
  const int tid = threadIdx.x;

  if (tid == 0) {
    int lo = 0, hi = N;                       // lower_bound(batch, g)
    while (lo < hi) { int mid = (lo + hi) >> 1; if (batch[mid] < g) lo = mid + 1; else hi = mid; }
    seg[0] = lo;
    hi = N;                                   // lower_bound(batch, g+1)
    while (lo < hi) { int mid = (lo + hi) >> 1; if (batch[mid] < g + 1) lo = mid + 1; else hi = mid; }
    seg[1] = lo;
  }
  __syncthreads();
  const int start = seg[0], end = seg[1];
  const int cnt = end - start;
  if (cnt <= 0) { sbuf[(size_t)g * HID + tid] = 0.f; return; }
  const bool fits = (cnt <= SCAP);

  // ---- phase 1: per-node scores; wave w handles nodes start+w, start+w+4, ...
  const int lane = tid & 31, wave = tid >> 5;
  const float4 q4 = ((const float4*)qk)[lane];
  for (int i = wave; i < cnt; i += 4) {
    const int n = start + i;
    const float4 xv = ((const float4*)(x + (size_t)n * HID))[lane];
    float d = xv.x * q4.x + xv.y * q4.y + xv.z * q4.z + xv.w * q4.w;
    for (int m = 16; m >= 1; m >>= 1) d += __shfl_xor(d, m, 32);
    if (lane == 0) {
      const float s = d * scale;
      if (fits) sc[i] = s; else attnbuf[n] = s;
    }
  }
  __syncthreads();

  // ---- phase 2: segment max
  float lm = -INFINITY;
  for (int i = tid; i < cnt; i += 128) lm = fmaxf(lm, fits ? sc[i] : attnbuf[start + i]);
  red[tid] = lm; __syncthreads();
  for (int s2 = 64; s2 >= 1; s2 >>= 1) { if (tid < s2) red[tid] = fmaxf(red[tid], red[tid + s2]); __syncthreads(); }
  const float gmax = red[0];
  __syncthreads();

  // ---- phase 3: exp + segment sum
  float ls = 0.f;
  for (int i = tid; i < cnt; i += 128) {
    const float e = __expf((fits ? sc[i] : attnbuf[start + i]) - gmax);
    if (fits) sc[i] = e; else attnbuf[start + i] = e;
    ls += e;
  }
  red[tid] = ls; __syncthreads();
  for (int s2 = 64; s2 >= 1; s2 >>= 1) { if (tid < s2) red[tid] += red[tid + s2]; __syncthreads(); }
  const float inv = 1.0f / red[0];
  __syncthreads();

  // ---- phase 4: s[g][tid] = sum_i attn[i] * x[start+i][tid]  (coalesced x reads)
  float acc = 0.f;
  for (int i = 0; i < cnt; ++i) {
    const float w = (fits ? sc[i] : attnbuf[start + i]) * inv;
    acc = fmaf(w, x[(size_t)(start + i) * HID + tid], acc);
  }
  sbuf[(size_t)g * HID + tid] = acc;
}

// out[m][n] = sum_k s[m][k]*Wc[n][k] + bc[n]  via V_WMMA_F32_16X16X4_F32.
// One wave per 16x16 tile. ISA layouts (32-bit A 16x4 / B 4x16):
//   lane L: row/col = L%16; lanes 0-15 carry K={k,k+1}, lanes 16-31 carry K={k+2,k+3}.
__global__ void out_gemm(const float* __restrict__ sbuf, const float* __restrict__ Wc,
                         const float* __restrict__ bc, float* __restrict__ out, int G) {
  const int lane = threadIdx.x;      // 0..31
  const int m0 = blockIdx.x * 16;
  const int n0 = blockIdx.y * 16;
  const int half = lane >> 4;
  const int l16 = lane & 15;
  int m = m0 + l16; if (m >= G) m = G - 1;   // clamp loads; stores are guarded
  const int n = n0 + l16;
  const float* arow = sbuf + (size_t)m * HID + 2 * half;
  const float* brow = Wc + (size_t)n * HID + 2 * half;
  v8f acc = {0.f, 0.f, 0.f, 0.f, 0.f, 0.f, 0.f, 0.f};
#pragma unroll
  for (int k = 0; k < HID; k += 4) {
    v2f a = *(const v2f*)(arow + k);
    v2f b = *(const v2f*)(brow + k);
    acc = __builtin_amdgcn_wmma_f32_16x16x4_f32(false, a, false, b, (short)0, acc,
                                                false, false);
  }
  const float bias = bc[n];
#pragma unroll
  for (int v = 0; v < 8; ++v) {
    const int row = m0 + v + 8 * half;   // C/D layout: VGPR v -> M = v + 8*(lane/16)
    if (row < G) out[(size_t)row * HID + n0 + l16] = acc[v] + bias;
  }
}

extern "C" void kernel_launch(void* const* d_in, const int* in_sizes, int n_in,
                              void* d_out, int out_size, void* d_ws, size_t ws_size,
                              hipStream_t stream) {
  const float* x     = (const float*)d_in[0];
  const int*   batch = (const int*)d_in[1];
  // d_in[2] = n_graphs (device scalar, unused: G derived from out_size)
  const float* query = (const float*)d_in[3];
  const float* Wk    = (const float*)d_in[4];
  // d_in[5] = bk : cancels in the segment softmax, unused
  const float* Wv    = (const float*)d_in[6];
  const float* bv    = (const float*)d_in[7];
  const float* Wo    = (const float*)d_in[8];
  const float* bo    = (const float*)d_in[9];

  const int H = in_sizes[3];          // 128
  const int N = in_sizes[1];          // 500000
  const int G = out_size / H;         // 5000
  const float scale = 1.0f / sqrtf((float)H);

  // Workspace layout (floats): attn fallback | s[G*H] | qk[H] | Wc[H*H] | bc[H]
  float* ws      = (float*)d_ws;
  float* attnbuf = ws;
  float* sbuf    = attnbuf + N;
  float* qk      = sbuf + (size_t)G * H;
  float* Wc      = qk + H;
  float* bc      = Wc + (size_t)H * H;

  setup_small<<<1, 128, 0, stream>>>(query, Wk, Wo, bv, bo, qk, bc);
  setup_wc<<<H, 128, 0, stream>>>(Wo, Wv, Wc);
  graph_pool<<<G, 128, 0, stream>>>(x, batch, qk, scale, attnbuf, sbuf, N, G);
  out_gemm<<<dim3((G + 15) / 16, H / 16), 32, 0, stream>>>(sbuf, Wc, bc, (float*)d_out, G);
}